// HOG_Embedding_68796786147740
// MI455X (gfx1250) — compile-verified
//
#include <hip/hip_runtime.h>
#include <math.h>

#define K_NN   20
#define NPTS   2048
#define NBATCH 8
#define NBINS  9
#define BNN    (NBATCH * NPTS)

typedef __attribute__((ext_vector_type(2))) float v2f;
typedef __attribute__((ext_vector_type(8))) float v8f;

__device__ __forceinline__ float pymod(float a, float b) {
  return a - b * floorf(a / b);   // python-style mod, matches jnp.mod
}

__device__ __forceinline__ unsigned lds_byte_off(const void* p) {
  // generic LDS address: low 32 bits are the LDS byte address used by
  // DS/async ops (ISA 10.2 aperture rules)
  return (unsigned)(uintptr_t)p;
}

// ---------------------------------------------------------------------------
// 1. KNN: per batch, per point, top-20 by neg squared distance (ref formula)
// ---------------------------------------------------------------------------
__global__ void knn_kernel(const float* __restrict__ X, int* __restrict__ idx) {
  __shared__ float sx[NPTS], sy[NPTS], sz[NPTS];
  const int b = blockIdx.y;
  const float* Xb = X + (size_t)b * 3 * NPTS;
  for (int i = threadIdx.x; i < NPTS; i += blockDim.x) {
    sx[i] = Xb[i];
    sy[i] = Xb[NPTS + i];
    sz[i] = Xb[2 * NPTS + i];
  }
  __syncthreads();
  const int p = blockIdx.x * blockDim.x + threadIdx.x;
  const float px = sx[p], py = sy[p], pz = sz[p];
  const float xxp = px * px + py * py + pz * pz;
  float val[K_NN];
  int   ind[K_NN];
#pragma unroll
  for (int t = 0; t < K_NN; ++t) { val[t] = -INFINITY; ind[t] = 0; }
  for (int j = 0; j < NPTS; ++j) {
    const float qx = sx[j], qy = sy[j], qz = sz[j];
    const float inner = px * qx + py * qy + pz * qz;
    const float xxq = qx * qx + qy * qy + qz * qz;
    const float nd = 2.0f * inner - xxp - xxq;
    if (nd > val[K_NN - 1]) {
      int t = K_NN - 1;
      while (t > 0 && val[t - 1] < nd) {   // strict: stable, lower index first on ties
        val[t] = val[t - 1]; ind[t] = ind[t - 1]; --t;
      }
      val[t] = nd; ind[t] = j;
    }
  }
  int* op = idx + ((size_t)b * NPTS + p) * K_NN;
#pragma unroll
  for (int t = 0; t < K_NN; ++t) op[t] = ind[t];
}

// ---------------------------------------------------------------------------
// 2. Per-point covariance -> top eigenpair (SVD surrogate)
// ---------------------------------------------------------------------------
__global__ void eig_kernel(const float* __restrict__ X, const int* __restrict__ idx,
                           float* __restrict__ grad, float* __restrict__ mag) {
  const int i = blockIdx.x * blockDim.x + threadIdx.x;
  if (i >= BNN) return;
  const int b = i / NPTS;
  const float* Xb = X + (size_t)b * 3 * NPTS;
  const int* ip = idx + (size_t)i * K_NN;
  float xs[K_NN], ys[K_NN], zs[K_NN];
  float mx = 0.f, my = 0.f, mz = 0.f;
#pragma unroll
  for (int t = 0; t < K_NN; ++t) {
    const int j = ip[t];
    xs[t] = Xb[j]; ys[t] = Xb[NPTS + j]; zs[t] = Xb[2 * NPTS + j];
    mx += xs[t]; my += ys[t]; mz += zs[t];
  }
  mx *= (1.0f / K_NN); my *= (1.0f / K_NN); mz *= (1.0f / K_NN);
  float c00 = 0, c01 = 0, c02 = 0, c11 = 0, c12 = 0, c22 = 0;
#pragma unroll
  for (int t = 0; t < K_NN; ++t) {
    const float dx = xs[t] - mx, dy = ys[t] - my, dz = zs[t] - mz;
    c00 += dx * dx; c01 += dx * dy; c02 += dx * dz;
    c11 += dy * dy; c12 += dy * dz; c22 += dz * dz;
  }
  const float q  = (c00 + c11 + c22) * (1.0f / 3.0f);
  const float p1 = c01 * c01 + c02 * c02 + c12 * c12;
  const float p2 = (c00 - q) * (c00 - q) + (c11 - q) * (c11 - q) +
                   (c22 - q) * (c22 - q) + 2.0f * p1;
  float eig;
  if (p2 <= 1e-30f) {
    eig = q;
  } else {
    const float pp = sqrtf(p2 * (1.0f / 6.0f));
    const float inv = 1.0f / pp;
    const float b00 = (c00 - q) * inv, b01 = c01 * inv, b02 = c02 * inv;
    const float b11 = (c11 - q) * inv, b12 = c12 * inv, b22 = (c22 - q) * inv;
    float r = 0.5f * (b00 * (b11 * b22 - b12 * b12) -
                      b01 * (b01 * b22 - b12 * b02) +
                      b02 * (b01 * b12 - b11 * b02));
    r = fminf(1.0f, fmaxf(-1.0f, r));
    eig = q + 2.0f * pp * cosf(acosf(r) * (1.0f / 3.0f));
  }
  const float r0x = c00 - eig, r0y = c01, r0z = c02;
  const float r1x = c01, r1y = c11 - eig, r1z = c12;
  const float r2x = c02, r2y = c12, r2z = c22 - eig;
  const float ax = r0y * r1z - r0z * r1y, ay = r0z * r1x - r0x * r1z, az = r0x * r1y - r0y * r1x;
  const float bx = r0y * r2z - r0z * r2y, by = r0z * r2x - r0x * r2z, bz = r0x * r2y - r0y * r2x;
  const float cx = r1y * r2z - r1z * r2y, cy = r1z * r2x - r1x * r2z, cz = r1x * r2y - r1y * r2x;
  const float na = ax * ax + ay * ay + az * az;
  const float nb = bx * bx + by * by + bz * bz;
  const float nc = cx * cx + cy * cy + cz * cz;
  float vx, vy, vz, nn;
  if (na >= nb && na >= nc)      { vx = ax; vy = ay; vz = az; nn = na; }
  else if (nb >= nc)             { vx = bx; vy = by; vz = bz; nn = nb; }
  else                           { vx = cx; vy = cy; vz = cz; nn = nc; }
  if (nn < 1e-30f) { vx = 1.f; vy = 0.f; vz = 0.f; nn = 1.f; }
  const float inv = rsqrtf(nn);
  vx *= inv; vy *= inv; vz *= inv;
  const float fax = fabsf(vx), fay = fabsf(vy), faz = fabsf(vz);
  const float comp = (fax >= fay && fax >= faz) ? vx : ((fay >= faz) ? vy : vz);
  if (comp < 0.f) { vx = -vx; vy = -vy; vz = -vz; }
  grad[(size_t)i * 3 + 0] = vx;
  grad[(size_t)i * 3 + 1] = vy;
  grad[(size_t)i * 3 + 2] = vz;
  mag[i] = sqrtf(sqrtf(fmaxf(eig, 0.0f)));
}

// ---------------------------------------------------------------------------
// 3. HOG histogram -> normalized -> raw-reshape layout (B, 18, N)
// ---------------------------------------------------------------------------
__global__ void hist_kernel(const int* __restrict__ idx, const float* __restrict__ grad,
                            const float* __restrict__ mag, float* __restrict__ h0) {
  const int i = blockIdx.x * blockDim.x + threadIdx.x;
  if (i >= BNN) return;
  const int b = i / NPTS, n = i % NPTS;
  const int* ip = idx + (size_t)i * K_NN;
  float hist[NBINS][2];
#pragma unroll
  for (int j = 0; j < NBINS; ++j) { hist[j][0] = 0.f; hist[j][1] = 0.f; }
  for (int t = 0; t < K_NN; ++t) {
    const int j = ip[t];
    const size_t gi = (size_t)b * NPTS + j;
    const float gx = grad[gi * 3], gy = grad[gi * 3 + 1], gz = grad[gi * 3 + 2];
    float m = mag[gi];
    if (m < 0.0f) m += 180.0f;
    const float zen = truncf(acosf(gz) * (180.0f / (float)M_PI));
    const float azi = truncf(atanf(gy / gx) * (180.0f / (float)M_PI));
    const float angs[2] = {zen, azi};
#pragma unroll
    for (int c = 0; c < 2; ++c) {
      float a = angs[c];
      if (a < 0.0f) a += 180.0f;
      const float bin = pymod(floorf(a * (1.0f / 20.0f) - 0.5f), 9.0f);
      const float fc = 20.0f * (pymod(bin + 1.0f, 9.0f) + 0.5f);
      const float fv = m * pymod(fc - a, 180.0f) * (1.0f / 20.0f);
      const float sc = 20.0f * (bin + 0.5f);
      const float sv = m * pymod(a - sc, 180.0f) * (1.0f / 20.0f);
      int ib = (int)bin;
      if (ib < 0) ib = 0;
      if (ib > NBINS - 1) ib = NBINS - 1;
      hist[ib][c] += fv;
      hist[(ib + 1) % NBINS][c] += sv;
    }
  }
#pragma unroll
  for (int c = 0; c < 2; ++c) {
    float s = 0.f;
#pragma unroll
    for (int j = 0; j < NBINS; ++j) s += hist[j][c] * hist[j][c];
    const float nrm = fmaxf(sqrtf(s), 1e-12f);
    const float inv = 1.0f / nrm;
#pragma unroll
    for (int j = 0; j < NBINS; ++j) hist[j][c] *= inv;
  }
  float* op = h0 + (size_t)b * 18 * NPTS + (size_t)n * 18;
#pragma unroll
  for (int j = 0; j < NBINS; ++j) { op[j * 2] = hist[j][0]; op[j * 2 + 1] = hist[j][1]; }
}

// ---------------------------------------------------------------------------
// 4. conv1x1 GEMM: 256 threads = 8 waves -> 32(M) x 256(N) tile.
//    Double-buffered 16x256 fp32 X panels staged via GLOBAL_LOAD_ASYNC_TO_LDS
//    (ASYNCcnt) overlapped with V_WMMA_F32_16X16X4_F32 compute.
//    Y[b,m,n] = sum_c W[m,c] * X[b,c,n] + bias[m]
// ---------------------------------------------------------------------------
#define NB 256                    // N columns per block
#define KB 16                     // K rows staged per panel
__global__ void gemm_wmma_kernel(const float* __restrict__ X, const float* __restrict__ W,
                                 const float* __restrict__ bias, float* __restrict__ Y,
                                 int C, int M) {
  __shared__ float sB[2][KB][NB];      // 2 x 16KB double-buffered panels
  const int lane = threadIdx.x & 31;
  const int wv   = threadIdx.x >> 5;   // 0..7
  const int wm   = wv >> 2;            // 0..1 : m-tile within block
  const int wn   = wv & 3;             // 0..3 : 64-col group within block
  const int half = lane >> 4;          // 0: K pair {0,1}; 1: K pair {2,3}
  const int li   = lane & 15;          // A: row m ; B: col n (within 16)
  const int nbase = blockIdx.x * NB;
  const int m0    = blockIdx.y * 32 + wm * 16;
  const int b     = blockIdx.z;
  const float* Xb = X + (size_t)b * C * NPTS;
  const float* Wr = W + (size_t)(m0 + li) * C;
  const unsigned ldsbase = lds_byte_off(&sB[0][0][0]);

  // per-thread staging coordinates: 4 float4 slots of the 16x256 panel
  const int row0 = (int)threadIdx.x >> 6;          // 0..3 (then +4 per u)
  const int col0 = ((int)threadIdx.x & 63) << 2;   // 0..252 step 4

  auto stage = [&](int k, int bufsel) {
    const unsigned dbase = ldsbase + (unsigned)bufsel * (KB * NB * 4);
    if (k + KB <= C) {
      // clean full panel: 4 unconditional async b128 per thread
#pragma unroll
      for (int u = 0; u < 4; ++u) {
        const int row = row0 + u * 4;
        const float* src = Xb + (size_t)(k + row) * NPTS + nbase + col0;
        const unsigned dst = dbase + (unsigned)((row * NB + col0) * 4);
        asm volatile("global_load_async_to_lds_b128 %0, %1, off"
                     :: "v"(dst), "v"(src) : "memory");
      }
    } else {
      // guarded tail panel: zero-pad rows beyond C
#pragma unroll
      for (int u = 0; u < 4; ++u) {
        const int row = row0 + u * 4;
        if (k + row < C) {
          const float* src = Xb + (size_t)(k + row) * NPTS + nbase + col0;
          const unsigned dst = dbase + (unsigned)((row * NB + col0) * 4);
          asm volatile("global_load_async_to_lds_b128 %0, %1, off"
                       :: "v"(dst), "v"(src) : "memory");
        } else {
          sB[bufsel][row][col0]     = 0.f;
          sB[bufsel][row][col0 + 1] = 0.f;
          sB[bufsel][row][col0 + 2] = 0.f;
          sB[bufsel][row][col0 + 3] = 0.f;
        }
      }
    }
  };

  v8f acc0 = {}, acc1 = {}, acc2 = {}, acc3 = {};
  const int P = (C + KB - 1) / KB;     // number of K panels

  stage(0, 0);
  for (int p = 0; p < P; ++p) {
    const int buf = p & 1;
    const int k = p * KB;
    // publish panel p: each wave waits on its own async loads, then barrier
    asm volatile("s_wait_asynccnt 0x0" ::: "memory");
    __syncthreads();
    // start staging panel p+1 into the other buffer (flies during compute)
    if (p + 1 < P) stage((p + 1) * KB, buf ^ 1);

    const bool full = (k + KB <= C);
#pragma unroll
    for (int i4 = 0; i4 < 4; ++i4) {
      const int kl = i4 * 4 + half * 2;   // local K row pair for this lane
      const int kg = k + kl;
      v2f a;
      if (full) {
        a.x = Wr[kg];
        a.y = Wr[kg + 1];
      } else {
        a.x = (kg < C)     ? Wr[kg]     : 0.f;
        a.y = (kg + 1 < C) ? Wr[kg + 1] : 0.f;
      }
      v2f b0, b1, b2, b3;
      b0.x = sB[buf][kl][wn * 64 + 0 * 16 + li];  b0.y = sB[buf][kl + 1][wn * 64 + 0 * 16 + li];
      b1.x = sB[buf][kl][wn * 64 + 1 * 16 + li];  b1.y = sB[buf][kl + 1][wn * 64 + 1 * 16 + li];
      b2.x = sB[buf][kl][wn * 64 + 2 * 16 + li];  b2.y = sB[buf][kl + 1][wn * 64 + 2 * 16 + li];
      b3.x = sB[buf][kl][wn * 64 + 3 * 16 + li];  b3.y = sB[buf][kl + 1][wn * 64 + 3 * 16 + li];
      acc0 = __builtin_amdgcn_wmma_f32_16x16x4_f32(false, a, false, b0, (short)0, acc0, false, false);
      acc1 = __builtin_amdgcn_wmma_f32_16x16x4_f32(false, a, false, b1, (short)0, acc1, false, false);
      acc2 = __builtin_amdgcn_wmma_f32_16x16x4_f32(false, a, false, b2, (short)0, acc2, false, false);
      acc3 = __builtin_amdgcn_wmma_f32_16x16x4_f32(false, a, false, b3, (short)0, acc3, false, false);
    }
    __syncthreads();   // all reads of this buffer done before it is restaged
  }

  // ---- write out: C/D layout: VGPR r -> row r (+8 for lanes 16-31) ----
  float* Yb = Y + (size_t)b * M * NPTS;
#pragma unroll
  for (int r = 0; r < 8; ++r) {
    const int m = m0 + r + half * 8;
    const float bi = bias[m];
    float* yrow = Yb + (size_t)m * NPTS + nbase + wn * 64 + li;
    yrow[0]  = acc0[r] + bi;
    yrow[16] = acc1[r] + bi;
    yrow[32] = acc2[r] + bi;
    yrow[48] = acc3[r] + bi;
  }
}

// ---------------------------------------------------------------------------
// 5. Per-channel mean/var over (B, N)
// ---------------------------------------------------------------------------
__global__ void stats_kernel(const float* __restrict__ T, int M,
                             float* __restrict__ mean, float* __restrict__ var) {
  const int m = blockIdx.x;
  __shared__ float ss[256], sq[256];
  float s = 0.f, q = 0.f;
  for (int i = threadIdx.x; i < BNN; i += blockDim.x) {
    const int b = i / NPTS, n = i % NPTS;
    const float v = T[((size_t)b * M + m) * NPTS + n];
    s += v; q += v * v;
  }
  ss[threadIdx.x] = s; sq[threadIdx.x] = q;
  __syncthreads();
  for (int off = 128; off > 0; off >>= 1) {
    if ((int)threadIdx.x < off) {
      ss[threadIdx.x] += ss[threadIdx.x + off];
      sq[threadIdx.x] += sq[threadIdx.x + off];
    }
    __syncthreads();
  }
  if (threadIdx.x == 0) {
    const float mn = ss[0] * (1.0f / BNN);
    mean[m] = mn;
    var[m] = sq[0] * (1.0f / BNN) - mn * mn;
  }
}

// ---------------------------------------------------------------------------
// 6. BN + leaky relu (0.2)
// ---------------------------------------------------------------------------
__global__ void bn_lrelu_kernel(const float* __restrict__ T, int M,
                                const float* __restrict__ mean, const float* __restrict__ var,
                                const float* __restrict__ g, const float* __restrict__ beta,
                                float* __restrict__ O) {
  const size_t i = (size_t)blockIdx.x * blockDim.x + threadIdx.x;
  if (i >= (size_t)NBATCH * M * NPTS) return;
  const int m = (int)((i / NPTS) % M);
  const float v = (T[i] - mean[m]) * rsqrtf(var[m] + 1e-5f) * g[m] + beta[m];
  O[i] = (v >= 0.0f) ? v : 0.2f * v;
}

// ---------------------------------------------------------------------------
// 7. BN(t2) + BN(ts) -> add -> leaky relu (0.2)
// ---------------------------------------------------------------------------
__global__ void bn2_add_lrelu_kernel(const float* __restrict__ T2, const float* __restrict__ TS,
                                     int M,
                                     const float* __restrict__ m2, const float* __restrict__ v2,
                                     const float* __restrict__ g2, const float* __restrict__ b2,
                                     const float* __restrict__ ms, const float* __restrict__ vs,
                                     const float* __restrict__ gs, const float* __restrict__ bs,
                                     float* __restrict__ O) {
  const size_t i = (size_t)blockIdx.x * blockDim.x + threadIdx.x;
  if (i >= (size_t)NBATCH * M * NPTS) return;
  const int m = (int)((i / NPTS) % M);
  const float a = (T2[i] - m2[m]) * rsqrtf(v2[m] + 1e-5f) * g2[m] + b2[m];
  const float s = (TS[i] - ms[m]) * rsqrtf(vs[m] + 1e-5f) * gs[m] + bs[m];
  const float v = a + s;
  O[i] = (v >= 0.0f) ? v : 0.2f * v;
}

// ---------------------------------------------------------------------------
// Host side
// ---------------------------------------------------------------------------
struct LayerP {
  const float *W1, *b1, *g1, *be1, *W2, *b2, *g2, *be2, *Ws, *bs, *gs, *bes;
};

static void run_layer(const float* in, int Cin, int Cmid, int Cout, const LayerP& P,
                      float* t1, float* act, float* t2, float* ts, float* out,
                      float* mean1, float* var1, float* mean2, float* var2,
                      float* means, float* vars, hipStream_t stream) {
  const dim3 blk(256);
  const dim3 g1(NPTS / NB, Cmid / 32, NBATCH);
  gemm_wmma_kernel<<<g1, blk, 0, stream>>>(in, P.W1, P.b1, t1, Cin, Cmid);
  stats_kernel<<<Cmid, 256, 0, stream>>>(t1, Cmid, mean1, var1);
  {
    const size_t tot = (size_t)NBATCH * Cmid * NPTS;
    bn_lrelu_kernel<<<(unsigned)((tot + 255) / 256), 256, 0, stream>>>(
        t1, Cmid, mean1, var1, P.g1, P.be1, act);
  }
  const dim3 g2d(NPTS / NB, Cout / 32, NBATCH);
  gemm_wmma_kernel<<<g2d, blk, 0, stream>>>(act, P.W2, P.b2, t2, Cmid, Cout);
  stats_kernel<<<Cout, 256, 0, stream>>>(t2, Cout, mean2, var2);
  gemm_wmma_kernel<<<g2d, blk, 0, stream>>>(in, P.Ws, P.bs, ts, Cin, Cout);
  stats_kernel<<<Cout, 256, 0, stream>>>(ts, Cout, means, vars);
  {
    const size_t tot = (size_t)NBATCH * Cout * NPTS;
    bn2_add_lrelu_kernel<<<(unsigned)((tot + 255) / 256), 256, 0, stream>>>(
        t2, ts, Cout, mean2, var2, P.g2, P.be2, means, vars, P.gs, P.bes, out);
  }
}

extern "C" void kernel_launch(void* const* d_in, const int* in_sizes, int n_in,
                              void* d_out, int out_size, void* d_ws, size_t ws_size,
                              hipStream_t stream) {
  const float* x;
  LayerP L1, L2;
  auto F = [&](int i) { return (const float*)d_in[i]; };

  if (n_in >= 25 && in_sizes[0] == NBATCH * 3 * NPTS) {
    // insertion-order flattening: x, L1{W1,b1,g1,be1,W2,b2,g2,be2,Ws,bs,gs,bes}, L2{...}
    x = F(0);
    int o = 1;
    LayerP* Ls[2] = {&L1, &L2};
    for (int l = 0; l < 2; ++l) {
      LayerP& L = *Ls[l];
      L.W1 = F(o + 0);  L.b1 = F(o + 1);  L.g1 = F(o + 2);  L.be1 = F(o + 3);
      L.W2 = F(o + 4);  L.b2 = F(o + 5);  L.g2 = F(o + 6);  L.be2 = F(o + 7);
      L.Ws = F(o + 8);  L.bs = F(o + 9);  L.gs = F(o + 10); L.bes = F(o + 11);
      o += 12;
    }
  } else {
    // sorted-pytree flattening: L1 sorted{W1,W2,Ws,b1,b2,be1,be2,bes,bs,g1,g2,gs}, L2 sorted, x
    int o = 0;
    LayerP* Ls[2] = {&L1, &L2};
    for (int l = 0; l < 2; ++l) {
      LayerP& L = *Ls[l];
      L.W1 = F(o + 0);  L.W2 = F(o + 1);  L.Ws = F(o + 2);
      L.b1 = F(o + 3);  L.b2 = F(o + 4);
      L.be1 = F(o + 5); L.be2 = F(o + 6); L.bes = F(o + 7); L.bs = F(o + 8);
      L.g1 = F(o + 9);  L.g2 = F(o + 10); L.gs = F(o + 11);
      o += 12;
    }
    x = F(24);
  }

  // workspace carve-out (floats, 256B-aligned blocks)
  size_t off = 0;
  auto alloc = [&](size_t nf) {
    float* p = (float*)d_ws + off;
    off += (nf + 63) & ~(size_t)63;
    return p;
  };
  int*   idx  = (int*)alloc((size_t)BNN * K_NN);
  float* grad = alloc((size_t)BNN * 3);
  float* mag  = alloc((size_t)BNN);
  float* h0   = alloc((size_t)NBATCH * 18 * NPTS);
  float* bufA = alloc((size_t)NBATCH * 256 * NPTS);   // t1 (<=256 ch)
  float* bufB = alloc((size_t)NBATCH * 256 * NPTS);   // act (<=256 ch)
  float* bufC = alloc((size_t)NBATCH * 512 * NPTS);   // t2 (<=512 ch)
  float* bufD = alloc((size_t)NBATCH * 512 * NPTS);   // ts (<=512 ch)
  float* out1 = alloc((size_t)NBATCH * 128 * NPTS);   // layer1 output
  float* mean1 = alloc(512); float* var1 = alloc(512);
  float* mean2 = alloc(512); float* var2 = alloc(512);
  float* means = alloc(512); float* vars = alloc(512);
  (void)ws_size;

  // HOG front-end
  knn_kernel<<<dim3(NPTS / 256, NBATCH), 256, 0, stream>>>(x, idx);
  eig_kernel<<<(BNN + 255) / 256, 256, 0, stream>>>(x, idx, grad, mag);
  hist_kernel<<<(BNN + 255) / 256, 256, 0, stream>>>(idx, grad, mag, h0);

  // residual MLP blocks
  run_layer(h0, 18, 64, 128, L1, bufA, bufB, bufC, bufD, out1,
            mean1, var1, mean2, var2, means, vars, stream);
  run_layer(out1, 128, 256, 512, L2, bufA, bufB, bufC, bufD, (float*)d_out,
            mean1, var1, mean2, var2, means, vars, stream);
  (void)out_size;
}